// GATLayer_22909355557526
// MI455X (gfx1250) — compile-verified
//
#include <hip/hip_runtime.h>
#include <hip/hip_bf16.h>

#define N_NODES 10000
#define N_EDGES 640000
#define N_GRAPHS 64
#define DD 128
#define NEG_SLOPE 0.01f

typedef __attribute__((ext_vector_type(2))) float v2f;
typedef __attribute__((ext_vector_type(8))) float v8f;

// ---------- helpers: monotonic float<->uint encoding for atomic max ----------
__device__ __forceinline__ unsigned enc_f32(float f) {
    unsigned u = __float_as_uint(f);
    return (u & 0x80000000u) ? ~u : (u | 0x80000000u);
}
__device__ __forceinline__ float dec_f32(unsigned e) {
    return (e & 0x80000000u) ? __uint_as_float(e & 0x7FFFFFFFu)
                             : __uint_as_float(~e);
}

// ---------- K0: zero-init m_enc, denom, phis ----------
__global__ void k_init(unsigned* __restrict__ menc, float* __restrict__ denom,
                       float* __restrict__ phis) {
    int i = blockIdx.x * blockDim.x + threadIdx.x;
    if (i < 2 * N_NODES) { menc[i] = 0u; denom[i] = 0.0f; }
    if (i < N_GRAPHS * 2 * DD) phis[i] = 0.0f;
}

// ---------- K1: feat = x@W^T (+bias) via V_WMMA_F32_16X16X4_F32 ----------
// grid: (ceil(10000/64), 2 blocks, 2 {fc,res}); block: 128 threads (4 waves).
// Each wave computes a 16-row x 128-col strip. Weights staged transposed in LDS.
__global__ void k_gemm(const float* __restrict__ x,
                       const float* __restrict__ Wfc, const float* __restrict__ bfc,
                       const float* __restrict__ Wres,
                       float* __restrict__ feat_ws, float* __restrict__ hout) {
    __shared__ float WT[DD * DD];  // WT[k][j] = W[j][k], 64KB
    const int b     = blockIdx.y;
    const int which = blockIdx.z;  // 0 = fc, 1 = res
    const float* W = (which == 0 ? Wfc : Wres) + (size_t)b * DD * DD;

    for (int idx = threadIdx.x; idx < DD * DD; idx += blockDim.x) {
        int j = idx >> 7, k = idx & 127;
        WT[k * DD + j] = W[idx];          // global read contiguous per thread-run
    }
    __syncthreads();

    const int lane = threadIdx.x & 31;
    const int wave = threadIdx.x >> 5;
    const int rowBase = blockIdx.x * 64 + wave * 16;
    const int l16 = lane & 15;
    const int hi  = lane >> 4;            // 0: lanes 0-15, 1: lanes 16-31

    // C/D layout: c[t][v] holds row (v + 8*hi), col (t*16 + l16)
    v8f c[8];
#pragma unroll
    for (int t = 0; t < 8; ++t) {
        float bv = (which == 0) ? bfc[b * DD + t * 16 + l16] : 0.0f;
#pragma unroll
        for (int v = 0; v < 8; ++v) c[t][v] = bv;
    }

    const int arow = min(rowBase + l16, N_NODES - 1);
    const float* xrow = x + (size_t)arow * DD;

    for (int k0 = 0; k0 < DD; k0 += 4) {
        const int krow = k0 + 2 * hi;
        // A 16x4 f32 layout: comp0 = K=krow, comp1 = K=krow+1 (8B aligned)
        v2f a = *(const v2f*)(xrow + krow);
#pragma unroll
        for (int t = 0; t < 8; ++t) {
            const int j = t * 16 + l16;
            v2f bf;
            bf.x = WT[krow * DD + j];        // lanes 0-15 consecutive -> no bank conflict
            bf.y = WT[(krow + 1) * DD + j];
            c[t] = __builtin_amdgcn_wmma_f32_16x16x4_f32(
                false, a, false, bf, (short)0, c[t], false, false);
        }
    }

#pragma unroll
    for (int t = 0; t < 8; ++t) {
        const int col = t * 16 + l16;
#pragma unroll
        for (int v = 0; v < 8; ++v) {
            const int row = rowBase + v + 8 * hi;
            if (row < N_NODES) {
                if (which == 0)
                    feat_ws[(size_t)b * N_NODES * DD + (size_t)row * DD + col] = c[t][v];
                else
                    hout[(size_t)row * (2 * DD) + b * DD + col] = c[t][v];
            }
        }
    }
}

// ---------- K2: a_self = feat@wl, a_nb = feat@wr (wave per row) ----------
__global__ void k_attn_coef(const float* __restrict__ feat,
                            const float* __restrict__ wl, const float* __restrict__ wr,
                            float* __restrict__ a_self, float* __restrict__ a_nb) {
    const int b = blockIdx.y;
    const int row = blockIdx.x * (blockDim.x >> 5) + (threadIdx.x >> 5);
    if (row >= N_NODES) return;
    const int lane = threadIdx.x & 31;
    float4 f = ((const float4*)(feat + (size_t)b * N_NODES * DD + (size_t)row * DD))[lane];
    float4 l = ((const float4*)(wl + b * DD))[lane];
    float4 r = ((const float4*)(wr + b * DD))[lane];
    float sl = f.x * l.x + f.y * l.y + f.z * l.z + f.w * l.w;
    float sr = f.x * r.x + f.y * r.y + f.z * r.z + f.w * r.w;
#pragma unroll
    for (int m = 16; m >= 1; m >>= 1) {
        sl += __shfl_xor(sl, m, 32);
        sr += __shfl_xor(sr, m, 32);
    }
    if (lane == 0) {
        a_self[b * N_NODES + row] = sl;
        a_nb[b * N_NODES + row]   = sr;
    }
}

__device__ __forceinline__ float edge_score(const float* a_self, const float* a_nb,
                                            int b, int s, int d) {
    float sc = a_self[b * N_NODES + d] + a_nb[b * N_NODES + s];
    return sc >= 0.0f ? sc : NEG_SLOPE * sc;
}

// ---------- K3: segment max over incoming edges ----------
__global__ void k_edge_max(const int* __restrict__ esrc, const int* __restrict__ edst,
                           const float* __restrict__ a_self, const float* __restrict__ a_nb,
                           unsigned* __restrict__ menc) {
    const int e = blockIdx.x * blockDim.x + threadIdx.x;
    const int b = blockIdx.y;
    if (e >= N_EDGES) return;
    float sc = edge_score(a_self, a_nb, b, esrc[e], edst[e]);
    atomicMax(&menc[b * N_NODES + edst[e]], enc_f32(sc));
}

// ---------- K3b: softmax denominator ----------
__global__ void k_edge_denom(const int* __restrict__ esrc, const int* __restrict__ edst,
                             const float* __restrict__ a_self, const float* __restrict__ a_nb,
                             const unsigned* __restrict__ menc, float* __restrict__ denom) {
    const int e = blockIdx.x * blockDim.x + threadIdx.x;
    const int b = blockIdx.y;
    if (e >= N_EDGES) return;
    const int d = edst[e];
    float sc = edge_score(a_self, a_nb, b, esrc[e], d);
    float ex = __expf(sc - dec_f32(menc[b * N_NODES + d]));
    atomicAdd(&denom[b * N_NODES + d], ex);
}

// ---------- K4: weighted neighbor aggregation (wave per edge) ----------
__global__ void k_edge_accum(const int* __restrict__ esrc, const int* __restrict__ edst,
                             const float* __restrict__ a_self, const float* __restrict__ a_nb,
                             const unsigned* __restrict__ menc, const float* __restrict__ denom,
                             const float* __restrict__ feat, float* __restrict__ hout) {
    const int b = blockIdx.y;
    const int lane = threadIdx.x & 31;
    const int wave = (blockIdx.x * blockDim.x + threadIdx.x) >> 5;
    const int nwaves = (gridDim.x * blockDim.x) >> 5;
    for (int e = wave; e < N_EDGES; e += nwaves) {
        const int s = esrc[e], d = edst[e];
        float sc = edge_score(a_self, a_nb, b, s, d);
        float w = __expf(sc - dec_f32(menc[b * N_NODES + d])) / denom[b * N_NODES + d];
        float4 f = ((const float4*)(feat + (size_t)b * N_NODES * DD + (size_t)s * DD))[lane];
        float* dst = hout + (size_t)d * (2 * DD) + b * DD + lane * 4;
        atomicAdd(dst + 0, w * f.x);
        atomicAdd(dst + 1, w * f.y);
        atomicAdd(dst + 2, w * f.z);
        atomicAdd(dst + 3, w * f.w);
    }
}

// ---------- K5: fused ReLU (in place on h) + per-graph sum pooling ----------
__global__ void k_relu_pool(float* __restrict__ hout, const int* __restrict__ gids,
                            float* __restrict__ phis) {
    const int idx = blockIdx.x * blockDim.x + threadIdx.x;  // over N_NODES * 64 float4s
    const int n = idx >> 6, q = idx & 63;
    if (n >= N_NODES) return;
    const int g = gids[n];
    float4 v = ((float4*)(hout + (size_t)n * (2 * DD)))[q];
    v.x = fmaxf(v.x, 0.0f); v.y = fmaxf(v.y, 0.0f);
    v.z = fmaxf(v.z, 0.0f); v.w = fmaxf(v.w, 0.0f);
    ((float4*)(hout + (size_t)n * (2 * DD)))[q] = v;
    float* p = phis + (size_t)g * (2 * DD) + q * 4;
    atomicAdd(p + 0, v.x); atomicAdd(p + 1, v.y);
    atomicAdd(p + 2, v.z); atomicAdd(p + 3, v.w);
}

extern "C" void kernel_launch(void* const* d_in, const int* in_sizes, int n_in,
                              void* d_out, int out_size, void* d_ws, size_t ws_size,
                              hipStream_t stream) {
    const float* x        = (const float*)d_in[0];
    const int*   esrc     = (const int*)d_in[1];
    const int*   edst     = (const int*)d_in[2];
    const int*   gids     = (const int*)d_in[3];
    const float* Wfc      = (const float*)d_in[4];
    const float* bfc      = (const float*)d_in[5];
    const float* wl       = (const float*)d_in[6];
    const float* wr       = (const float*)d_in[7];
    const float* Wres     = (const float*)d_in[8];

    float* h    = (float*)d_out;                         // [10000, 256]
    float* phis = (float*)d_out + (size_t)N_NODES * 2 * DD; // [64, 256]

    // workspace layout
    char* ws = (char*)d_ws;
    float*    feat   = (float*)ws;                                  // 2*10000*128 f32
    float*    a_self = (float*)(ws + (size_t)2 * N_NODES * DD * 4);
    float*    a_nb   = a_self + 2 * N_NODES;
    unsigned* menc   = (unsigned*)(a_nb + 2 * N_NODES);
    float*    denom  = (float*)(menc + 2 * N_NODES);

    // K0: init
    k_init<<<dim3((2 * N_NODES + 255) / 256), dim3(256), 0, stream>>>(menc, denom, phis);

    // K1: GEMMs (feat+bias -> ws, res -> h)
    k_gemm<<<dim3((N_NODES + 63) / 64, 2, 2), dim3(128), 0, stream>>>(
        x, Wfc, bfc, Wres, feat, h);

    // K2: attention coefficients
    k_attn_coef<<<dim3((N_NODES + 3) / 4, 2), dim3(128), 0, stream>>>(
        feat, wl, wr, a_self, a_nb);

    // K3: segment max
    k_edge_max<<<dim3((N_EDGES + 255) / 256, 2), dim3(256), 0, stream>>>(
        esrc, edst, a_self, a_nb, menc);

    // K3b: softmax denominator
    k_edge_denom<<<dim3((N_EDGES + 255) / 256, 2), dim3(256), 0, stream>>>(
        esrc, edst, a_self, a_nb, menc, denom);

    // K4: weighted aggregation (wave per edge)
    k_edge_accum<<<dim3(2048, 2), dim3(256), 0, stream>>>(
        esrc, edst, a_self, a_nb, menc, denom, feat, h);

    // K5: ReLU + pooling
    k_relu_pool<<<dim3((N_NODES * 64 + 255) / 256), dim3(256), 0, stream>>>(
        h, gids, phis);
}